// RMPP_63891933495324
// MI455X (gfx1250) — compile-verified
//
#include <hip/hip_runtime.h>
#include <hip/hip_bf16.h>

// ---------------------------------------------------------------------------
// Kalman-smoother scan for MI455X (gfx1250, wave32, WMMA f32 16x16x4).
//
// fwd:  sequential r=0..R-1, one wave. Stores C_r, W_r=L_r^{-1}, u_r.
//       Uses V_WMMA_F32_16X16X4_F32 for C = -(W*MtP)^T and P = Ptr - C*C^T.
// bwd vs: one wave, 1-row chain (g = us), stores vs.
// bwd ws: 4 waves, each a 16x32 state tile; per step two 16x32x32 GEMMs
//         entirely on f32 WMMA:  Z = G + (-VW)*C ;  VW' = Z*W ;
//         out[r] = VW' + vs[r].
// ---------------------------------------------------------------------------

typedef float v2f __attribute__((ext_vector_type(2)));
typedef float v8f __attribute__((ext_vector_type(8)));

__device__ __forceinline__ v8f wmma4(v2f a, v2f b, v8f c) {
  // D = A(16x4,f32) * B(4x16,f32) + C(16x16,f32)
  return __builtin_amdgcn_wmma_f32_16x16x4_f32(false, a, false, b,
                                               (short)0, c, false, false);
}

// A fragment from a 32xK row-major LDS matrix, row tile s (rows 16s..16s+15),
// K-chunk q (cols 4q..4q+3). ISA layout: lanes 0-15 -> K={4q,4q+1},
// lanes 16-31 -> K={4q+2,4q+3}, M = lane&15.
__device__ __forceinline__ v2f afrag_lds(const float* A, int q, int lane,
                                         int s, bool neg) {
  const int lo = lane & 15, hi = lane >> 4;
  const int m = 16 * s + lo;
  const int k0 = 4 * q + 2 * hi;
  v2f a;
  a.x = A[m * 32 + k0];
  a.y = A[m * 32 + k0 + 1];
  if (neg) { a.x = -a.x; a.y = -a.y; }
  return a;
}

// B fragment (4x16) from row-major LDS matrix B[k][n] (or B = M^T if trans).
__device__ __forceinline__ v2f bfrag_lds(const float* B, int q, int lane,
                                         int t, bool trans) {
  const int lo = lane & 15, hi = lane >> 4;
  const int n = 16 * t + lo;
  const int k0 = 4 * q + 2 * hi;
  v2f b;
  if (!trans) { b.x = B[k0 * 32 + n];  b.y = B[(k0 + 1) * 32 + n]; }
  else        { b.x = B[n * 32 + k0];  b.y = B[n * 32 + k0 + 1];   }
  return b;
}

// Out(32x32) = [negOut] ( (negA?-A:A) * (btrans?B^T:B) + C0 ), LDS->LDS.
__device__ __forceinline__ void gemm32_lds(const float* A, bool negA,
                                           const float* B, bool btrans,
                                           const float* C0,
                                           float* Out, bool otrans, bool negOut,
                                           int lane) {
  const int lo = lane & 15, hi = lane >> 4;
  for (int s = 0; s < 2; ++s) {
    for (int t = 0; t < 2; ++t) {
      v8f acc;
      for (int v = 0; v < 8; ++v)
        acc[v] = C0 ? C0[(16 * s + v + 8 * hi) * 32 + 16 * t + lo] : 0.0f;
      for (int q = 0; q < 8; ++q) {
        v2f a = afrag_lds(A, q, lane, s, negA);
        v2f b = bfrag_lds(B, q, lane, t, btrans);
        acc = wmma4(a, b, acc);
      }
      for (int v = 0; v < 8; ++v) {
        const int m = 16 * s + v + 8 * hi, n = 16 * t + lo;
        const float val = negOut ? -acc[v] : acc[v];
        if (otrans) Out[n * 32 + m] = val;
        else        Out[m * 32 + n] = val;
      }
    }
  }
}

// ---------------------------------------------------------------------------
// Forward scan: one wave marches r = 0..R-1.
// ---------------------------------------------------------------------------
__global__ void fwd_kernel(const float* __restrict__ H,   // R x 32 x 32
                           const float* __restrict__ G,   // R x 32
                           const float* __restrict__ Mt,  // 32 x 32 (x_trans_mat)
                           const float* __restrict__ Ptr_,// 32 x 32 (x_trans_prec)
                           const float* __restrict__ P0,  // 32 x 32 (x_init_prec)
                           float* __restrict__ Cs, float* __restrict__ Ws,
                           float* __restrict__ us, int R) {
  __shared__ __align__(16) float Plds[1024], Alds[1024], Llds[1024];
  __shared__ __align__(16) float Wlds[1024], Clds[1024];
  __shared__ __align__(16) float MtPl[1024], MtPMtl[1024], Mtl[1024], Ptl[1024];
  __shared__ float ul[32], zl[32];
  const int lane = threadIdx.x;

  for (int j = 0; j < 32; ++j) {
    Mtl[lane * 32 + j]  = Mt[lane * 32 + j];
    Ptl[lane * 32 + j]  = Ptr_[lane * 32 + j];
    Plds[lane * 32 + j] = P0[lane * 32 + j];
    Clds[lane * 32 + j] = 0.0f;
  }
  ul[lane] = 0.0f;
  __syncthreads();
  // MtP = Mt * Ptr ; MtPMt = MtP * Mt^T   (lane = row)
  for (int j = 0; j < 32; ++j) {
    float s = 0.f;
    for (int k = 0; k < 32; ++k) s += Mtl[lane * 32 + k] * Ptl[k * 32 + j];
    MtPl[lane * 32 + j] = s;
  }
  __syncthreads();
  for (int j = 0; j < 32; ++j) {
    float s = 0.f;
    for (int k = 0; k < 32; ++k) s += MtPl[lane * 32 + k] * Mtl[j * 32 + k];
    MtPMtl[lane * 32 + j] = s;
  }
  __syncthreads();

  for (int r = 0; r < R; ++r) {
    if (r + 1 < R)  // pull next H tile toward L2/L0 (global_prefetch_b8)
      __builtin_prefetch(&H[(size_t)(r + 1) * 1024 + lane * 32], 0, 1);
    const float m = (r < R - 1) ? 1.0f : 0.0f;

    // A = P + H_r + m*MtPMt   (lane = row, contiguous per-lane loads)
    for (int j = 0; j < 32; ++j)
      Alds[lane * 32 + j] = Plds[lane * 32 + j] +
                            H[(size_t)r * 1024 + lane * 32 + j] +
                            m * MtPMtl[lane * 32 + j];
    __syncthreads();

    // Cholesky A = L*L^T (lower), lane = row
    for (int j = 0; j < 32; ++j) {
      const float d = sqrtf(Alds[j * 32 + j]);
      const float lij = (lane >= j) ? Alds[lane * 32 + j] / d : 0.0f;
      Llds[lane * 32 + j] = lij;
      __syncthreads();
      if (lane > j)
        for (int k = j + 1; k <= lane; ++k)
          Alds[lane * 32 + k] -= lij * Llds[k * 32 + j];
      __syncthreads();
    }

    // W = L^{-1} (lower triangular inverse), lane = column (independent)
    {
      const int c = lane;
      for (int i = 0; i < 32; ++i) {
        float wi = 0.0f;
        if (i >= c) {
          float s = (i == c) ? 1.0f : 0.0f;
          for (int k = c; k < i; ++k) s -= Llds[i * 32 + k] * Wlds[k * 32 + c];
          wi = s / Llds[i * 32 + i];
        }
        Wlds[i * 32 + c] = wi;
      }
    }
    __syncthreads();

    // u' = (g - u*C_old^T) * W^T   (uses OLD C, per reference ordering)
    {
      float z = G[(size_t)r * 32 + lane];
      for (int k = 0; k < 32; ++k) z -= ul[k] * Clds[lane * 32 + k];
      zl[lane] = z;
      __syncthreads();
      float un = 0.f;
      for (int k = 0; k < 32; ++k) un += zl[k] * Wlds[lane * 32 + k];
      __syncthreads();
      ul[lane] = un;
      us[(size_t)r * 32 + lane] = un;
    }
    __syncthreads();

    // C = -(W * MtP)^T        [WMMA f32]
    gemm32_lds(Wlds, false, MtPl, false, nullptr, Clds, true, true, lane);
    __syncthreads();

    // spill C_r, W_r (row-major, 128B per lane, b128 stores)
    {
      float4* cd = (float4*)(Cs + (size_t)r * 1024 + lane * 32);
      float4* wd = (float4*)(Ws + (size_t)r * 1024 + lane * 32);
      const float4* csrc = (const float4*)(Clds + lane * 32);
      const float4* wsrc = (const float4*)(Wlds + lane * 32);
      for (int v = 0; v < 8; ++v) { cd[v] = csrc[v]; wd[v] = wsrc[v]; }
    }

    // P = Ptr - C*C^T         [WMMA f32: A=-C, B=C^T, acc init = Ptr]
    gemm32_lds(Clds, true, Clds, true, Ptl, Plds, false, false, lane);
    __syncthreads();
  }
}

// ---------------------------------------------------------------------------
// Backward "vs" chain: single row, g = us.  vw' = (g - vw*C) * W
// ---------------------------------------------------------------------------
__global__ void bwd_vs_kernel(const float* __restrict__ Cs,
                              const float* __restrict__ Ws,
                              const float* __restrict__ us,
                              float* __restrict__ vs, int R) {
  __shared__ float vwb[32], zb[32];
  const int lane = threadIdx.x;
  float vw = 0.0f;
  for (int r = R - 1; r >= 0; --r) {
    const float* Cg = Cs + (size_t)r * 1024;
    const float* Wg = Ws + (size_t)r * 1024;
    __syncthreads();
    vwb[lane] = vw;
    __syncthreads();
    float z = us[(size_t)r * 32 + lane];
    for (int k = 0; k < 32; ++k) z -= vwb[k] * Cg[k * 32 + lane];
    zb[lane] = z;
    __syncthreads();
    float nv = 0.f;
    for (int k = 0; k < 32; ++k) nv += zb[k] * Wg[k * 32 + lane];
    vw = nv;
    vs[(size_t)r * 32 + lane] = nv;
  }
}

// ---------------------------------------------------------------------------
// Backward Monte chains: 4 blocks x 16 rows. Hot loop = 32 WMMAs/step.
// ---------------------------------------------------------------------------
__global__ void bwd_ws_kernel(const float* __restrict__ Cs,
                              const float* __restrict__ Ws,
                              const float* __restrict__ vs,
                              const float* __restrict__ eps,  // R x 64 x 32
                              float* __restrict__ out, int R) {
  __shared__ __align__(16) float zbuf[16 * 32];
  const int lane = threadIdx.x;
  const int blk = blockIdx.x;  // rows 16*blk .. 16*blk+15 of the 64 monte rows
  const int lo = lane & 15, hi = lane >> 4;

  v8f vw0, vw1;  // VW tile, cols 0-15 / 16-31, WMMA C/D layout
  for (int v = 0; v < 8; ++v) { vw0[v] = 0.f; vw1[v] = 0.f; }

  for (int r = R - 1; r >= 0; --r) {
    const float* Cg = Cs + (size_t)r * 1024;
    const float* Wg = Ws + (size_t)r * 1024;
    __syncthreads();
    // spill -VW (D layout -> row-major) so it can be re-read as A fragments
    for (int v = 0; v < 8; ++v) {
      const int row = v + 8 * hi;
      zbuf[row * 32 + lo]      = -vw0[v];
      zbuf[row * 32 + 16 + lo] = -vw1[v];
    }
    __syncthreads();

    // Z = G + (-VW)*C_r   (acc initialized from eps tile)
    v8f z0, z1;
    for (int v = 0; v < 8; ++v) {
      const size_t base = ((size_t)r * 64 + 16 * blk + v + 8 * hi) * 32;
      z0[v] = eps[base + lo];
      z1[v] = eps[base + 16 + lo];
    }
    for (int q = 0; q < 8; ++q) {
      const int k0 = 4 * q + 2 * hi;
      v2f a; a.x = zbuf[lo * 32 + k0]; a.y = zbuf[lo * 32 + k0 + 1];
      v2f b0, b1;
      b0.x = Cg[k0 * 32 + lo];       b0.y = Cg[(k0 + 1) * 32 + lo];
      b1.x = Cg[k0 * 32 + 16 + lo];  b1.y = Cg[(k0 + 1) * 32 + 16 + lo];
      z0 = wmma4(a, b0, z0);
      z1 = wmma4(a, b1, z1);
    }
    __syncthreads();
    for (int v = 0; v < 8; ++v) {
      const int row = v + 8 * hi;
      zbuf[row * 32 + lo]      = z0[v];
      zbuf[row * 32 + 16 + lo] = z1[v];
    }
    __syncthreads();

    // VW' = Z * W_r
    v8f n0, n1;
    for (int v = 0; v < 8; ++v) { n0[v] = 0.f; n1[v] = 0.f; }
    for (int q = 0; q < 8; ++q) {
      const int k0 = 4 * q + 2 * hi;
      v2f a; a.x = zbuf[lo * 32 + k0]; a.y = zbuf[lo * 32 + k0 + 1];
      v2f b0, b1;
      b0.x = Wg[k0 * 32 + lo];       b0.y = Wg[(k0 + 1) * 32 + lo];
      b1.x = Wg[k0 * 32 + 16 + lo];  b1.y = Wg[(k0 + 1) * 32 + 16 + lo];
      n0 = wmma4(a, b0, n0);
      n1 = wmma4(a, b1, n1);
    }
    vw0 = n0; vw1 = n1;

    // out[r] = VW' + vs[r]  (vs broadcast over rows)
    const float vsv0 = vs[(size_t)r * 32 + lo];
    const float vsv1 = vs[(size_t)r * 32 + 16 + lo];
    for (int v = 0; v < 8; ++v) {
      const size_t base = ((size_t)r * 64 + 16 * blk + v + 8 * hi) * 32;
      out[base + lo]      = vw0[v] + vsv0;
      out[base + 16 + lo] = vw1[v] + vsv1;
    }
  }
}

extern "C" void kernel_launch(void* const* d_in, const int* in_sizes, int n_in,
                              void* d_out, int out_size, void* d_ws, size_t ws_size,
                              hipStream_t stream) {
  const float* H    = (const float*)d_in[0];  // x_hessian_diags R*32*32
  const float* G    = (const float*)d_in[1];  // x_grads         R*32
  const float* Mt   = (const float*)d_in[2];  // x_trans_mat     32*32
  const float* Ptr_ = (const float*)d_in[3];  // x_trans_prec    32*32
  const float* P0   = (const float*)d_in[4];  // x_init_prec     32*32
  const float* eps  = (const float*)d_in[5];  // epsx            R*64*32
  const int R = in_sizes[1] / 32;

  float* ws = (float*)d_ws;
  float* Cs = ws;                        // R*1024
  float* Ws = Cs + (size_t)R * 1024;     // R*1024
  float* us = Ws + (size_t)R * 1024;     // R*32
  float* vs = us + (size_t)R * 32;       // R*32
  float* out = (float*)d_out;

  fwd_kernel<<<1, 32, 0, stream>>>(H, G, Mt, Ptr_, P0, Cs, Ws, us, R);
  bwd_vs_kernel<<<1, 32, 0, stream>>>(Cs, Ws, us, vs, R);
  bwd_ws_kernel<<<4, 32, 0, stream>>>(Cs, Ws, vs, eps, out, R);
}